// NwKernelModel_30339648979316
// MI455X (gfx1250) — compile-verified
//
#include <hip/hip_runtime.h>
#include <math.h>

// Problem constants (from reference): B=512, N_BG=10000, F=32, H=128
#define BATCH 512
#define NBG   10000
#define FDIM  32
#define HDIM  128

typedef __attribute__((ext_vector_type(2))) float v2f;
typedef __attribute__((ext_vector_type(8))) float v8f;

__device__ __forceinline__ v8f wmma_f32_16x16x4(v2f a, v2f b, v8f c) {
    // (neg_a, A, neg_b, B, c_mod, C, reuse_a, reuse_b)
    return __builtin_amdgcn_wmma_f32_16x16x4_f32(false, a, false, b, (short)0, c,
                                                 false, false);
}

// ---------------------------------------------------------------------------
// Kernel 1: full MLP for a 16-row tile of X, producing |b_vec| tile (f32 WMMA).
// grid = 32 blocks (512/16), block = 256 threads = 8 waves.
// Wave w computes the 16x16 tile at columns [16w, 16w+16) of each hidden layer.
// ---------------------------------------------------------------------------
__global__ __launch_bounds__(256) void mlp_babs_kernel(
    const float* __restrict__ X,
    const float* __restrict__ W0, const float* __restrict__ b0,
    const float* __restrict__ W1, const float* __restrict__ b1,
    const float* __restrict__ Wout, const float* __restrict__ bout,
    float* __restrict__ babs)
{
    __shared__ float h0s[16][HDIM + 4];   // +4 pad: kill LDS bank conflicts
    __shared__ float h1s[16][HDIM + 4];

    const int tid  = threadIdx.x;
    const int wv   = tid >> 5;        // wave id 0..7
    const int lane = tid & 31;
    const int half = lane >> 4;       // 0 for lanes 0-15, 1 for lanes 16-31
    const int l15  = lane & 15;
    const int r0   = blockIdx.x * 16; // batch-row tile base

    // ---------------- layer 0: H0 = relu(X @ W0 + b0) ----------------
    {
        const int c0 = wv * 16;
        v8f acc = {0.f,0.f,0.f,0.f,0.f,0.f,0.f,0.f};
        #pragma unroll
        for (int k0 = 0; k0 < FDIM; k0 += 4) {
            const int ka = k0 + 2 * half;
            v2f a, b;
            a.x = X[(r0 + l15) * FDIM + ka];
            a.y = X[(r0 + l15) * FDIM + ka + 1];
            b.x = W0[ka * HDIM + c0 + l15];
            b.y = W0[(ka + 1) * HDIM + c0 + l15];
            acc = wmma_f32_16x16x4(a, b, acc);
        }
        const float bias = b0[c0 + l15];
        #pragma unroll
        for (int v = 0; v < 8; ++v) {
            const int m = v + 8 * half;
            const float val = acc[v] + bias;
            h0s[m][c0 + l15] = val > 0.f ? val : 0.f;
        }
    }
    __syncthreads();

    // ---------------- layer 1: H1 = relu(H0 @ W1 + b1) ----------------
    {
        const int c0 = wv * 16;
        v8f acc = {0.f,0.f,0.f,0.f,0.f,0.f,0.f,0.f};
        #pragma unroll 4
        for (int k0 = 0; k0 < HDIM; k0 += 4) {
            const int ka = k0 + 2 * half;
            v2f a, b;
            a.x = h0s[l15][ka];
            a.y = h0s[l15][ka + 1];
            b.x = W1[ka * HDIM + c0 + l15];
            b.y = W1[(ka + 1) * HDIM + c0 + l15];
            acc = wmma_f32_16x16x4(a, b, acc);
        }
        const float bias = b1[c0 + l15];
        #pragma unroll
        for (int v = 0; v < 8; ++v) {
            const int m = v + 8 * half;
            const float val = acc[v] + bias;
            h1s[m][c0 + l15] = val > 0.f ? val : 0.f;
        }
    }
    __syncthreads();

    // ---------------- layer 2: babs = |H1 @ Wout + bout| ----------------
    if (wv < 2) {   // F=32 -> only 2 column tiles; wave-uniform branch, EXEC all-1s
        const int c0 = wv * 16;
        v8f acc = {0.f,0.f,0.f,0.f,0.f,0.f,0.f,0.f};
        #pragma unroll 4
        for (int k0 = 0; k0 < HDIM; k0 += 4) {
            const int ka = k0 + 2 * half;
            v2f a, b;
            a.x = h1s[l15][ka];
            a.y = h1s[l15][ka + 1];
            b.x = Wout[ka * FDIM + c0 + l15];
            b.y = Wout[(ka + 1) * FDIM + c0 + l15];
            acc = wmma_f32_16x16x4(a, b, acc);
        }
        const float bias = bout[c0 + l15];
        #pragma unroll
        for (int v = 0; v < 8; ++v) {
            const int m = v + 8 * half;
            babs[(r0 + m) * FDIM + c0 + l15] = fabsf(acc[v] + bias);
        }
    }
}

// ---------------------------------------------------------------------------
// Kernel 2: Nadaraya-Watson. One block per batch row; 256 threads stride over
// the 10000 background points. x_bg rows read as float4 (global_load_b128);
// per-row X and |b_vec| live in per-thread registers (fully unrolled arrays).
// ---------------------------------------------------------------------------
__global__ __launch_bounds__(256) void nw_kernel(
    const float* __restrict__ X,
    const float* __restrict__ xbg,
    const float* __restrict__ ybg,
    const float* __restrict__ babs,
    float* __restrict__ out)
{
    __shared__ float xs[FDIM];
    __shared__ float bs[FDIM];
    __shared__ float pw[8];
    __shared__ float pwy[8];

    const int tid = threadIdx.x;
    const int b   = blockIdx.x;

    if (tid < FDIM) {
        xs[tid] = X[b * FDIM + tid];
        bs[tid] = babs[b * FDIM + tid];
    }
    __syncthreads();

    // Broadcast row data into registers (LDS reads are uniform -> cheap once).
    float xr[FDIM], br[FDIM];
    #pragma unroll
    for (int f = 0; f < FDIM; ++f) { xr[f] = xs[f]; br[f] = bs[f]; }

    float sw = 0.f, swy = 0.f;
    for (int n = tid; n < NBG; n += 256) {
        const float4* p = (const float4*)(xbg + n * FDIM);
        float d = 0.f;
        #pragma unroll
        for (int j = 0; j < FDIM / 4; ++j) {
            const float4 v = p[j];
            d += br[4 * j + 0] * fabsf(xr[4 * j + 0] - v.x);
            d += br[4 * j + 1] * fabsf(xr[4 * j + 1] - v.y);
            d += br[4 * j + 2] * fabsf(xr[4 * j + 2] - v.z);
            d += br[4 * j + 3] * fabsf(xr[4 * j + 3] - v.w);
        }
        const float w = __expf(-d);      // v_exp_f32 (transcendental pipe)
        sw  += w;
        swy += w * ybg[n];
    }

    // wave32 butterfly reduction
    #pragma unroll
    for (int off = 16; off > 0; off >>= 1) {
        sw  += __shfl_xor(sw,  off, 32);
        swy += __shfl_xor(swy, off, 32);
    }
    if ((tid & 31) == 0) { pw[tid >> 5] = sw; pwy[tid >> 5] = swy; }
    __syncthreads();

    if (tid == 0) {
        float tw = 0.f, twy = 0.f;
        #pragma unroll
        for (int i = 0; i < 8; ++i) { tw += pw[i]; twy += pwy[i]; }
        out[b] = twy / tw;
    }
}

// ---------------------------------------------------------------------------
extern "C" void kernel_launch(void* const* d_in, const int* in_sizes, int n_in,
                              void* d_out, int out_size, void* d_ws, size_t ws_size,
                              hipStream_t stream) {
    const float* X    = (const float*)d_in[0];
    const float* xbg  = (const float*)d_in[1];
    const float* ybg  = (const float*)d_in[2];
    const float* W0   = (const float*)d_in[3];
    const float* b0   = (const float*)d_in[4];
    const float* W1   = (const float*)d_in[5];
    const float* b1   = (const float*)d_in[6];
    const float* Wout = (const float*)d_in[7];
    const float* bout = (const float*)d_in[8];
    float* out  = (float*)d_out;
    float* babs = (float*)d_ws;           // 512*32*4 = 64 KB scratch

    mlp_babs_kernel<<<BATCH / 16, 256, 0, stream>>>(X, W0, b0, W1, b1, Wout, bout, babs);
    nw_kernel<<<BATCH, 256, 0, stream>>>(X, xbg, ybg, babs, out);
}